// TensorSplatter_31817117728919
// MI455X (gfx1250) — compile-verified
//
#include <hip/hip_runtime.h>
#include <hip/hip_bf16.h>

typedef float v2f __attribute__((ext_vector_type(2)));
typedef float v8f __attribute__((ext_vector_type(8)));

#define NA_MAX 2048
#define DD 64

__device__ __forceinline__ float wave_reduce_add(float v) {
    v += __shfl_xor(v, 16, 32);
    v += __shfl_xor(v, 8, 32);
    v += __shfl_xor(v, 4, 32);
    v += __shfl_xor(v, 2, 32);
    v += __shfl_xor(v, 1, 32);
    return v;
}

// ---------------------------------------------------------------------------
// Stage 1: per-atom projections via V_WMMA_F32_16X16X4_F32.
// One wave per atom. A (16x64, K swept 4 at a time) rows = the 16 feature
// slices; B (64x16) column j = the weight vector paired with feature j.
// diag(D) = the 16 projections. c0e/c0o via shfl reduction; biases folded.
// projT layout: [18][Na]  (row 0=c0e+biases, 1=c0o+biases, 2..4=Vo, 5..7=Ve,
//                          8..12=Te, 13..17=To)
// ---------------------------------------------------------------------------
__global__ __launch_bounds__(256) void atom_proj_kernel(
    const float* __restrict__ even_scalar, const float* __restrict__ odd_scalar,
    const float* __restrict__ odd_vector,  const float* __restrict__ even_vector,
    const float* __restrict__ even_tensor, const float* __restrict__ odd_tensor,
    const float* __restrict__ w_e0, const float* __restrict__ b_e0,
    const float* __restrict__ w_o0, const float* __restrict__ b_o0,
    const float* __restrict__ w_e1, const float* __restrict__ b_e1,
    const float* __restrict__ w_o1, const float* __restrict__ b_o1,
    const float* __restrict__ w_e2, const float* __restrict__ b_e2,
    const float* __restrict__ w_o2, const float* __restrict__ b_o2,
    float* __restrict__ projT, int Na)
{
    __shared__ float cbuf[8 * 256];
    const int wave = threadIdx.x >> 5;
    const int lane = threadIdx.x & 31;
    const int a    = blockIdx.x * 8 + wave;
    const int j     = lane & 15;   // A row == B column for this lane
    const int khalf = lane >> 4;   // 0: K0/K1, 1: K2/K3 within each 4-chunk

    // Per-lane A-row source (feature slice) and B-column weight vector.
    const float* fbase;
    int stride;
    if (j < 3)       { fbase = odd_vector  + (size_t)a * DD * 3 + j;        stride = 3; }
    else if (j < 6)  { fbase = even_vector + (size_t)a * DD * 3 + (j - 3);  stride = 3; }
    else if (j < 11) { fbase = even_tensor + (size_t)a * DD * 5 + (j - 6);  stride = 5; }
    else             { fbase = odd_tensor  + (size_t)a * DD * 5 + (j - 11); stride = 5; }
    const float* wcol = (j < 3) ? w_e1 : (j < 6) ? w_o1 : (j < 11) ? w_e2 : w_o2;

    v8f c = {};
    #pragma unroll
    for (int kb = 0; kb < 16; ++kb) {
        const int k0 = kb * 4 + khalf * 2;
        v2f A, B;
        A.x = fbase[(size_t)k0 * stride];
        A.y = fbase[(size_t)(k0 + 1) * stride];
        B.x = wcol[k0];
        B.y = wcol[k0 + 1];
        c = __builtin_amdgcn_wmma_f32_16x16x4_f32(false, A, false, B,
                                                  (short)0, c, false, false);
    }

    // Dump accumulator to LDS, pull the diagonal.
    const int wbase = wave * 256;
    #pragma unroll
    for (int r = 0; r < 8; ++r) cbuf[wbase + r * 32 + lane] = c[r];

    // c0e / c0o: 64-length dots via wave reduction (biases folded in).
    float pe = even_scalar[(size_t)a * DD + lane]      * w_e0[lane]
             + even_scalar[(size_t)a * DD + 32 + lane] * w_e0[32 + lane];
    float po = odd_scalar[(size_t)a * DD + lane]       * w_o0[lane]
             + odd_scalar[(size_t)a * DD + 32 + lane]  * w_o0[32 + lane];
    pe = wave_reduce_add(pe);
    po = wave_reduce_add(po);

    __syncthreads();

    if (lane < 16) {
        // diag(m): m<8 -> lane=m, reg=m ; m>=8 -> lane=m+16, reg=m-8
        const int pos = (j < 8) ? (j * 33) : ((j - 8) * 32 + j + 16);
        projT[(size_t)(2 + j) * Na + a] = cbuf[wbase + pos];
    }
    if (lane == 0) {
        projT[a]      = pe + b_e0[0] + b_e1[0] + b_e2[0];
        projT[Na + a] = po + b_o0[0] + b_o1[0] + b_o2[0];
    }
}

// ---------------------------------------------------------------------------
// Stage 2: pair-wise kernel. One block (8 wave32s) per query row.
// Pure streaming-store bound (~403 MB out) -> non-temporal stores.
// Weights row cached in LDS so normalized_weights needs no global re-read.
// ---------------------------------------------------------------------------
__global__ __launch_bounds__(256) void splat_kernel(
    const float* __restrict__ qc, const float* __restrict__ ac,
    const float* __restrict__ alpha, const float* __restrict__ projT,
    float* __restrict__ out_total, float* __restrict__ out_even,
    float* __restrict__ out_odd,   float* __restrict__ out_w,
    float* __restrict__ out_nw,    float* __restrict__ out_ws,
    float* __restrict__ out_alpha, float* __restrict__ out_dist,
    float* __restrict__ out_y1,    float* __restrict__ out_y2, int Na)
{
    __shared__ float wcache[NA_MAX];
    __shared__ float red[24];
    __shared__ float bcast[1];

    const int q   = blockIdx.x;
    const int tid = threadIdx.x;
    const float qx = qc[q * 3 + 0];
    const float qy = qc[q * 3 + 1];
    const float qz = qc[q * 3 + 2];
    const size_t rowbase = (size_t)q * Na;

    const float SQRT2 = 1.4142135623730951f;
    const float SQRT6 = 2.449489742783178f;

    float s_w = 0.f, s_e = 0.f, s_o = 0.f;

    for (int a = tid; a < Na; a += 256) {
        const float axv = ac[a * 3 + 0];
        const float ayv = ac[a * 3 + 1];
        const float azv = ac[a * 3 + 2];
        const float al  = alpha[a];

        const float rx = qx - axv, ry = qy - ayv, rz = qz - azv;
        const float d2 = rx * rx + ry * ry + rz * rz;
        const float dist = sqrtf(d2);
        const float w = expf(-al * dist * dist);

        const float dist_c = fmaxf(dist, 1e-8f);
        const float invd = 1.0f / dist_c;
        const bool gz = dist > 1e-6f;
        const float dx = gz ? rx * invd : 0.0f;
        const float dy = gz ? ry * invd : 0.0f;
        const float dz = gz ? rz * invd : 0.0f;
        const float n = sqrtf(dx * dx + dy * dy + dz * dz);
        const bool ngz = n > 1e-6f;
        const float invn = 1.0f / fmaxf(n, 1e-6f);
        const float y1x = ngz ? dx * invn : 0.0f;
        const float y1y = ngz ? dy * invn : 0.0f;
        const float y1z = ngz ? dz * invn : 0.0f;

        const float xx = y1x * y1x, yy = y1y * y1y, zz = y1z * y1z;
        const float t0 = (xx - yy) / SQRT2;
        const float t1 = (-xx - yy + 2.0f * zz) / SQRT6;
        const float t2 = SQRT2 * (y1x * y1y);
        const float t3 = SQRT2 * (y1x * y1z);
        const float t4 = SQRT2 * (y1y * y1z);

        const float e_term = projT[a]
            + y1x * projT[2 * Na + a] + y1y * projT[3 * Na + a] + y1z * projT[4 * Na + a]
            + t0 * projT[8 * Na + a]  + t1 * projT[9 * Na + a]  + t2 * projT[10 * Na + a]
            + t3 * projT[11 * Na + a] + t4 * projT[12 * Na + a];
        const float o_term = projT[Na + a]
            + y1x * projT[5 * Na + a] + y1y * projT[6 * Na + a] + y1z * projT[7 * Na + a]
            + t0 * projT[13 * Na + a] + t1 * projT[14 * Na + a] + t2 * projT[15 * Na + a]
            + t3 * projT[16 * Na + a] + t4 * projT[17 * Na + a];

        s_w += w;
        s_e += w * e_term;
        s_o += w * o_term;
        wcache[a] = w;

        const size_t p = rowbase + a;
        __builtin_nontemporal_store(w,    out_w + p);
        __builtin_nontemporal_store(al,   out_alpha + p);
        __builtin_nontemporal_store(dist, out_dist + p);
        __builtin_nontemporal_store(y1x,  out_y1 + p * 3 + 0);
        __builtin_nontemporal_store(y1y,  out_y1 + p * 3 + 1);
        __builtin_nontemporal_store(y1z,  out_y1 + p * 3 + 2);
        __builtin_nontemporal_store(t0,   out_y2 + p * 5 + 0);
        __builtin_nontemporal_store(t1,   out_y2 + p * 5 + 1);
        __builtin_nontemporal_store(t2,   out_y2 + p * 5 + 2);
        __builtin_nontemporal_store(t3,   out_y2 + p * 5 + 3);
        __builtin_nontemporal_store(t4,   out_y2 + p * 5 + 4);
    }

    s_w = wave_reduce_add(s_w);
    s_e = wave_reduce_add(s_e);
    s_o = wave_reduce_add(s_o);
    const int wave = tid >> 5, lane = tid & 31;
    if (lane == 0) { red[wave] = s_w; red[8 + wave] = s_e; red[16 + wave] = s_o; }
    __syncthreads();

    if (tid == 0) {
        float tw = 0.f, te = 0.f, td = 0.f;
        #pragma unroll
        for (int i = 0; i < 8; ++i) { tw += red[i]; te += red[8 + i]; td += red[16 + i]; }
        const float ws = fmaxf(tw, 1e-8f);
        const float ec = te / ws;
        const float oc = td / ws;
        out_total[q] = ec + oc;
        out_even[q]  = ec;
        out_odd[q]   = oc;
        out_ws[q]    = tw;     // raw sum per reference
        bcast[0] = ws;
    }
    __syncthreads();

    const float ws = bcast[0];
    for (int a = tid; a < Na; a += 256) {
        __builtin_nontemporal_store(wcache[a] / ws, out_nw + rowbase + a);
    }
}

extern "C" void kernel_launch(void* const* d_in, const int* in_sizes, int n_in,
                              void* d_out, int out_size, void* d_ws, size_t ws_size,
                              hipStream_t stream) {
    const float* qc          = (const float*)d_in[0];
    const float* ac          = (const float*)d_in[1];
    const float* alpha       = (const float*)d_in[2];
    const float* even_scalar = (const float*)d_in[3];
    const float* odd_scalar  = (const float*)d_in[4];
    const float* odd_vector  = (const float*)d_in[5];
    const float* even_vector = (const float*)d_in[6];
    const float* even_tensor = (const float*)d_in[7];
    const float* odd_tensor  = (const float*)d_in[8];
    const float* w_e0 = (const float*)d_in[9];
    const float* b_e0 = (const float*)d_in[10];
    const float* w_o0 = (const float*)d_in[11];
    const float* b_o0 = (const float*)d_in[12];
    const float* w_e1 = (const float*)d_in[13];
    const float* b_e1 = (const float*)d_in[14];
    const float* w_o1 = (const float*)d_in[15];
    const float* b_o1 = (const float*)d_in[16];
    const float* w_e2 = (const float*)d_in[17];
    const float* b_e2 = (const float*)d_in[18];
    const float* w_o2 = (const float*)d_in[19];
    const float* b_o2 = (const float*)d_in[20];

    const int Nq = in_sizes[0] / 3;
    const int Na = in_sizes[1] / 3;

    float* projT = (float*)d_ws;   // 18 * Na floats

    atom_proj_kernel<<<Na / 8, 256, 0, stream>>>(
        even_scalar, odd_scalar, odd_vector, even_vector, even_tensor, odd_tensor,
        w_e0, b_e0, w_o0, b_o0, w_e1, b_e1, w_o1, b_o1, w_e2, b_e2, w_o2, b_o2,
        projT, Na);

    const size_t NqNa = (size_t)Nq * Na;
    float* out       = (float*)d_out;
    float* out_total = out;
    float* out_even  = out + Nq;
    float* out_odd   = out + 2 * (size_t)Nq;
    float* out_w     = out + 3 * (size_t)Nq;
    float* out_nw    = out_w + NqNa;
    float* out_ws    = out_nw + NqNa;
    float* out_alpha = out_ws + Nq;
    float* out_dist  = out_alpha + NqNa;
    float* out_y1    = out_dist + NqNa;
    float* out_y2    = out_y1 + 3 * NqNa;

    splat_kernel<<<Nq, 256, 0, stream>>>(
        qc, ac, alpha, projT,
        out_total, out_even, out_odd, out_w, out_nw, out_ws,
        out_alpha, out_dist, out_y1, out_y2, Na);
}